// ImageMaskPointLoss_67053029425440
// MI455X (gfx1250) — compile-verified
//
#include <hip/hip_runtime.h>
#include <hip/hip_bf16.h>
#include <math.h>

// ---------------- problem constants (match reference) ----------------
#define FOCAL_ALPHA 0.25f
#define N_IMG 32
#define H_DIM 1024
#define W_DIM 1024
#define P_PTS 16
#define HW   (H_DIM * W_DIM)     // 1,048,576 elements per image
#define HW4  (HW / 4)            // 262,144 float4 per image

#define BLOCKS_PER_IMG 64
#define THREADS 256              // 8 wave32 per block
#define STRIDE  (BLOCKS_PER_IMG * THREADS)   // 16384 float4 per sweep
#define ITERS   (HW4 / STRIDE)               // exactly 16, no remainder

typedef float f32x4 __attribute__((ext_vector_type(4)));
typedef float v8f   __attribute__((ext_vector_type(8)));
typedef float v2f   __attribute__((ext_vector_type(2)));

// ---------------------------------------------------------------------
// Full-precision wave32 sum using V_WMMA_F32_16X16X4_F32.
// A[m][k]: lanes 0-15 supply (M=lane, K=0/1), lanes 16-31 (M=lane-16, K=2/3).
// With a0=a1=x and B=ones: D[m][n] = 2*(x(m) + x(m+16)) for every n.
// D rows 0-7 live in lanes 0-15 (8 VGPRs), rows 8-15 in lanes 16-31, so
// in-lane sum of the 8 D regs + shfl_xor(16) = 2 * (sum over all 32 lanes).
// EXEC is all ones at the call site (uniform control flow).
// ---------------------------------------------------------------------
__device__ __forceinline__ float wave_sum_wmma(float x) {
  v2f a; a[0] = x;    a[1] = x;
  v2f b; b[0] = 1.0f; b[1] = 1.0f;
  v8f c = {};
  c = __builtin_amdgcn_wmma_f32_16x16x4_f32(
      /*neg_a=*/false, a, /*neg_b=*/false, b,
      /*c_mod=*/(short)0, c, /*reuse_a=*/false, /*reuse_b=*/false);
  float s = ((c[0] + c[1]) + (c[2] + c[3])) + ((c[4] + c[5]) + (c[6] + c[7]));
  s += __shfl_xor(s, 16, 32);
  return 0.5f * s;
}

// ---------------------------------------------------------------------
// Pass 1: streaming fused sigmoid/softplus/focal + per-image dice sums.
// Exactly ITERS guard-free iterations per thread; NT b128 loads (single-use
// 256 MiB stream vs 192 MB L2) + speculative prefetch one stride ahead.
// ws layout: 4 floats per block at ws[4*(n*BLOCKS_PER_IMG + bx)]:
//   [0]=focal partial  [1]=sum(prob*t)  [2]=sum(prob)  [3]=sum(t)
// ---------------------------------------------------------------------
__global__ __launch_bounds__(THREADS)
void mask_loss_partial(const f32x4* __restrict__ pred,
                       const f32x4* __restrict__ gt,
                       float* __restrict__ ws) {
  const int n = blockIdx.y;
  const size_t base = (size_t)n * HW4 + (size_t)(blockIdx.x * THREADS + threadIdx.x);

  float facc = 0.0f, dacc = 0.0f, pacc = 0.0f, tacc = 0.0f;

#pragma unroll 4
  for (int it = 0; it < ITERS; ++it) {
    const size_t idx = base + (size_t)it * STRIDE;
    f32x4 z4 = __builtin_nontemporal_load(&pred[idx]);
    f32x4 t4 = __builtin_nontemporal_load(&gt[idx]);
    // speculative prefetch (TH=0): dropped silently if past the buffer end
    __builtin_prefetch(&pred[idx + STRIDE], 0, 0);
    __builtin_prefetch(&gt[idx + STRIDE], 0, 0);
#pragma unroll
    for (int k = 0; k < 4; ++k) {
      const float z = z4[k], t = t4[k];
      // shared transcendental core (3 trans ops/element, irreducible)
      const float e    = __expf(-fabsf(z));                 // exp(-|z|)
      const float inv  = __builtin_amdgcn_rcpf(1.0f + e);   // 1/(1+e) = sigmoid(|z|)
      const float pm   = e * inv;                           // sigmoid(-|z|)
      const float sp   = __logf(1.0f + e) + fmaxf(z, 0.0f); // softplus(z), stable
      const bool  pos  = (z >= 0.0f);
      const bool  tt   = (t != 0.0f);                       // t is exactly {0,1}
      const float prob = pos ? inv : pm;                    // sigmoid(z)
      const float om   = (tt == pos) ? pm : inv;            // 1 - p_t
      const float ce   = fmaf(-z, t, sp);                   // softplus(z) - z*t
      const float a_t  = fmaf(-0.5f, t, 0.75f);             // 0.25*t + 0.75*(1-t)
      facc  = fmaf(a_t * ce, om * om, facc);                // gamma == 2
      dacc  = fmaf(prob, t, dacc);
      pacc += prob;
      tacc += t;
    }
  }

  // wave32 reduction through the matrix pipe (f32 WMMA keeps full precision)
  float fw = wave_sum_wmma(facc);
  float dw = wave_sum_wmma(dacc);
  float pw = wave_sum_wmma(pacc);
  float tw = wave_sum_wmma(tacc);

  __shared__ float sw[THREADS / 32][4];
  const int lane = threadIdx.x & 31;
  const int wave = threadIdx.x >> 5;
  if (lane == 0) { sw[wave][0] = fw; sw[wave][1] = dw; sw[wave][2] = pw; sw[wave][3] = tw; }
  __syncthreads();
  if (threadIdx.x == 0) {
    float f = 0, d = 0, p = 0, t = 0;
#pragma unroll
    for (int w = 0; w < THREADS / 32; ++w) {
      f += sw[w][0]; d += sw[w][1]; p += sw[w][2]; t += sw[w][3];
    }
    float* o = ws + (size_t)4 * ((size_t)n * BLOCKS_PER_IMG + blockIdx.x);
    o[0] = f; o[1] = d; o[2] = p; o[3] = t;
  }
}

// ---------------------------------------------------------------------
// Pass 2: single block. Point-loss gather (1024 points), deterministic
// tree reduce, per-image dice combine, final scalar.
// ---------------------------------------------------------------------
__global__ __launch_bounds__(1024)
void finalize_loss(const float* __restrict__ pred,
                   const int* __restrict__ pos_pts,
                   const int* __restrict__ neg_pts,
                   const float* __restrict__ ws,
                   float* __restrict__ out) {
  __shared__ float sdata[1024];
  __shared__ float simg[N_IMG][2];   // [0]=dice term, [1]=focal partial
  const int tid = threadIdx.x;

  // ---- point loss: 512 pos + 512 neg BCE-with-logits samples ----
  {
    int which = tid >> 9;            // 0 = pos, 1 = neg
    int r = tid & 511;
    int n = r >> 4;                  // image
    int p = r & 15;                  // point
    const int* pts = which ? neg_pts : pos_pts;
    int x = pts[(n * P_PTS + p) * 2 + 0];
    int y = pts[(n * P_PTS + p) * 2 + 1];
    float z  = pred[(size_t)n * HW + (size_t)y * W_DIM + (size_t)x];
    float zz = which ? z : -z;       // pos target 1 -> softplus(-z); neg -> softplus(z)
    sdata[tid] = __logf(1.0f + __expf(-fabsf(zz))) + fmaxf(zz, 0.0f);
  }
  __syncthreads();
  for (int s = 512; s > 0; s >>= 1) {   // fixed-order tree: deterministic
    if (tid < s) sdata[tid] += sdata[tid + s];
    __syncthreads();
  }

  // ---- per-image combine of pass-1 partials (fixed order) ----
  if (tid < N_IMG) {
    const float* b = ws + (size_t)4 * (size_t)tid * BLOCKS_PER_IMG;
    float f = 0, d = 0, p = 0, t = 0;
    for (int j = 0; j < BLOCKS_PER_IMG; ++j) {
      f += b[4 * j + 0]; d += b[4 * j + 1]; p += b[4 * j + 2]; t += b[4 * j + 3];
    }
    simg[tid][0] = 1.0f - (2.0f * d + 1.0f) / (p + t + 1.0f);
    simg[tid][1] = f;
  }
  __syncthreads();

  if (tid == 0) {
    float dice = 0.0f, focal = 0.0f;
    for (int n = 0; n < N_IMG; ++n) { dice += simg[n][0]; focal += simg[n][1]; }
    dice  *= (1.0f / (float)N_IMG);
    focal *= (1.0f / ((float)N_IMG * (float)HW));
    float point = sdata[0] * (1.0f / (float)N_IMG);
    out[0] = dice + focal + point;   // W_MASK = W_POINTS = 1
  }
}

// ---------------------------------------------------------------------
extern "C" void kernel_launch(void* const* d_in, const int* in_sizes, int n_in,
                              void* d_out, int out_size, void* d_ws, size_t ws_size,
                              hipStream_t stream) {
  (void)in_sizes; (void)n_in; (void)out_size; (void)ws_size;
  const f32x4* pred4 = (const f32x4*)d_in[0];
  const f32x4* gt4   = (const f32x4*)d_in[1];
  const float* pred  = (const float*)d_in[0];
  const int*   pos   = (const int*)d_in[2];
  const int*   neg   = (const int*)d_in[3];
  float*       ws    = (float*)d_ws;          // needs 2048*4*4 = 32 KiB

  dim3 grid(BLOCKS_PER_IMG, N_IMG);           // 2048 blocks x 256 threads
  mask_loss_partial<<<grid, THREADS, 0, stream>>>(pred4, gt4, ws);
  finalize_loss<<<1, 1024, 0, stream>>>(pred, pos, neg, ws, (float*)d_out);
}